// LightGCNModel_2010044694695
// MI455X (gfx1250) — compile-verified
//
#include <hip/hip_runtime.h>

typedef __attribute__((ext_vector_type(2))) float v2f;
typedef __attribute__((ext_vector_type(8))) float v8f;

#define DIMS 64

// ---------------------------------------------------------------------------
// Zero a buffer with float4 stores.
// ---------------------------------------------------------------------------
__global__ void zero_f4(float4* __restrict__ p, long n4) {
  long i = (long)blockIdx.x * blockDim.x + threadIdx.x;
  if (i < n4) p[i] = make_float4(0.f, 0.f, 0.f, 0.f);
}

// ---------------------------------------------------------------------------
// Edge-parallel COO SpMM scatter: next[r,:] += v * cur[c,:]
// 16 lanes per edge, float4 per lane (one b128 load, 4 f32 atomics).
// FIRST=1: layer-0 embedding table is the virtual concat of user/item weights.
// Working set (cur 76.8MB + next 76.8MB) is L2-resident on MI455X (192MB L2),
// so the random gather + atomics stay on-chip; only edge arrays stream from HBM.
// ---------------------------------------------------------------------------
template <int FIRST>
__global__ void spmm_scatter(const float* __restrict__ vals,
                             const int* __restrict__ rows,
                             const int* __restrict__ cols,
                             const float* __restrict__ userW,
                             const float* __restrict__ itemW,
                             const float* __restrict__ cur,
                             float* __restrict__ nxt,
                             int n_edges, int n_users) {
  long t = (long)blockIdx.x * blockDim.x + threadIdx.x;
  long e = t >> 4;
  int p = (int)(t & 15);
  if (e >= n_edges) return;
  int r = rows[e];
  int c = cols[e];
  float v = vals[e];
  const float* src;
  if (FIRST) {
    src = (c < n_users) ? (userW + (long)c * DIMS)
                        : (itemW + (long)(c - n_users) * DIMS);
  } else {
    src = cur + (long)c * DIMS;
  }
  float4 x = *(const float4*)(src + 4 * p);
  float* dst = nxt + (long)r * DIMS + 4 * p;
#if defined(__HIP_DEVICE_COMPILE__)
  unsafeAtomicAdd(dst + 0, v * x.x);   // global_atomic_add_f32 (L2-resident)
  unsafeAtomicAdd(dst + 1, v * x.y);
  unsafeAtomicAdd(dst + 2, v * x.z);
  unsafeAtomicAdd(dst + 3, v * x.w);
#endif
}

// ---------------------------------------------------------------------------
// Gather-scale-accumulate via WMMA:  OUT(16x64) += X_gathered(16x64) * (0.25*I)
// One wave per 16-row output tile. For each 16-column block:
//   D = sum_k A_k(16x4) * B_k(4x16) + C,  B = 0.25 * identity chunk.
// MODE 0: X from embedding tables, C = 0 (initializes all of d_out).
// MODE 1: X from a layer buffer, C = current d_out tile.
// EXEC is all-ones for every participating wave (uniform early-exit only).
// ---------------------------------------------------------------------------
template <int MODE>
__global__ void gather_add_wmma(const float* __restrict__ userW,
                                const float* __restrict__ itemW,
                                const float* __restrict__ src,
                                const int* __restrict__ uidx,
                                const int* __restrict__ iidx,
                                float* __restrict__ out,
                                int n_user_out, int n_out_rows, int n_users) {
  const int lane = threadIdx.x & 31;
  const int wave = threadIdx.x >> 5;
  const int tile = blockIdx.x * (blockDim.x >> 5) + wave;
  if (tile * 16 >= n_out_rows) return;  // wave-uniform

  const int n  = lane & 15;  // A-matrix row M / B-and-C column N
  const int kk = lane >> 4;  // half-wave selector (K pair / row half)

  // Source row for A-matrix row M = n of this tile.
  const int i_row = tile * 16 + n;
  const int node = (i_row < n_user_out) ? uidx[i_row]
                                        : (n_users + iidx[i_row - n_user_out]);
  const float* arow;
  if (MODE == 0) {
    arow = (node < n_users) ? (userW + (long)node * DIMS)
                            : (itemW + (long)(node - n_users) * DIMS);
  } else {
    arow = src + (long)node * DIMS;
  }

  const float s = 0.25f;  // 1/(N_LAYERS+1), distributed across all partial sums

  for (int cb = 0; cb < 4; ++cb) {  // four 16-column blocks of d=64
    v8f c = {};
    if (MODE == 1) {
#pragma unroll
      for (int j = 0; j < 8; ++j)
        c[j] = out[(long)(tile * 16 + j + 8 * kk) * DIMS + cb * 16 + n];
    }
#pragma unroll
    for (int k = 0; k < 4; ++k) {
      // A: lane holds X[row n][cb*16 + 4k + 2kk + {0,1}] per ISA 16x4 layout
      const float* ap = arow + cb * 16 + 4 * k + 2 * kk;
      v2f a;
      a[0] = ap[0];
      a[1] = ap[1];
      // B: 0.25 * rows (4k..4k+3) of I_16
      const int k0 = 4 * k + 2 * kk;
      v2f b;
      b[0] = (n == k0) ? s : 0.f;
      b[1] = (n == k0 + 1) ? s : 0.f;
      c = __builtin_amdgcn_wmma_f32_16x16x4_f32(
          /*neg_a=*/false, a, /*neg_b=*/false, b,
          /*c_mod=*/(short)0, c, /*reuse_a=*/false, /*reuse_b=*/false);
    }
#pragma unroll
    for (int j = 0; j < 8; ++j)
      out[(long)(tile * 16 + j + 8 * kk) * DIMS + cb * 16 + n] = c[j];
  }
}

// ---------------------------------------------------------------------------
// Launcher
// ---------------------------------------------------------------------------
extern "C" void kernel_launch(void* const* d_in, const int* in_sizes, int n_in,
                              void* d_out, int out_size, void* d_ws, size_t ws_size,
                              hipStream_t stream) {
  (void)n_in; (void)ws_size;
  const float* userW = (const float*)d_in[0];
  const float* itemW = (const float*)d_in[1];
  const float* vals  = (const float*)d_in[2];
  const int*   rows  = (const int*)d_in[3];
  const int*   cols  = (const int*)d_in[4];
  const int*   uidx  = (const int*)d_in[5];
  const int*   iidx  = (const int*)d_in[6];
  float* out = (float*)d_out;

  const int n_users     = in_sizes[0] / DIMS;
  const int n_items     = in_sizes[1] / DIMS;
  const int n_edges     = in_sizes[2];
  const int n_nodes     = n_users + n_items;
  const int n_user_out  = in_sizes[5];
  const int n_out_rows  = out_size / DIMS;

  // Two ping-pong node buffers (each n_nodes * 64 f32 = 76.8MB).
  float* W0 = (float*)d_ws;
  float* W1 = W0 + (size_t)n_nodes * DIMS;

  const long n4      = (long)n_nodes * DIMS / 4;
  const int  zblocks = (int)((n4 + 255) / 256);
  const long sthr    = (long)n_edges * 16;
  const int  sblocks = (int)((sthr + 255) / 256);
  const int  tiles   = (n_out_rows + 15) / 16;
  const int  gblocks = (tiles + 7) / 8;  // 8 waves (tiles) per 256-thread block

  // out = 0.25 * emb[gathered]   (writes every element of d_out)
  gather_add_wmma<0><<<gblocks, 256, 0, stream>>>(
      userW, itemW, nullptr, uidx, iidx, out, n_user_out, n_out_rows, n_users);

  // Layer 1: W0 = A @ emb ; out += 0.25 * W0[gathered]
  zero_f4<<<zblocks, 256, 0, stream>>>((float4*)W0, n4);
  spmm_scatter<1><<<sblocks, 256, 0, stream>>>(
      vals, rows, cols, userW, itemW, nullptr, W0, n_edges, n_users);
  gather_add_wmma<1><<<gblocks, 256, 0, stream>>>(
      userW, itemW, W0, uidx, iidx, out, n_user_out, n_out_rows, n_users);

  // Layer 2: W1 = A @ W0 ; out += 0.25 * W1[gathered]
  zero_f4<<<zblocks, 256, 0, stream>>>((float4*)W1, n4);
  spmm_scatter<0><<<sblocks, 256, 0, stream>>>(
      vals, rows, cols, userW, itemW, W0, W1, n_edges, n_users);
  gather_add_wmma<1><<<gblocks, 256, 0, stream>>>(
      userW, itemW, W1, uidx, iidx, out, n_user_out, n_out_rows, n_users);

  // Layer 3: W0 = A @ W1 ; out += 0.25 * W0[gathered]
  zero_f4<<<zblocks, 256, 0, stream>>>((float4*)W0, n4);
  spmm_scatter<0><<<sblocks, 256, 0, stream>>>(
      vals, rows, cols, userW, itemW, W1, W0, n_edges, n_users);
  gather_add_wmma<1><<<gblocks, 256, 0, stream>>>(
      userW, itemW, W0, uidx, iidx, out, n_user_out, n_out_rows, n_users);
}